// FeatureTextGraphBotSAI_55774445306294
// MI455X (gfx1250) — compile-verified
//
#include <hip/hip_runtime.h>
#include <math.h>

#define SLOPE 0.01f

typedef float v2f __attribute__((ext_vector_type(2)));
typedef float v8f __attribute__((ext_vector_type(8)));

__device__ __forceinline__ float leaky(float x) { return x >= 0.f ? x : SLOPE * x; }

__device__ __forceinline__ v8f wmma4(v2f a, v2f b, v8f c) {
  // D(16x16,f32) = A(16x4,f32) * B(4x16,f32) + C   (V_WMMA_F32_16X16X4_F32)
  return __builtin_amdgcn_wmma_f32_16x16x4_f32(false, a, false, b, (short)0, c, false, false);
}

__device__ __forceinline__ void atomicMaxF(float* addr, float v) {
  if (v >= 0.f) atomicMax((int*)addr, __float_as_int(v));
  else          atomicMin((unsigned int*)addr, __float_as_uint(v));
}

// ---------------------------------------------------------------------------
// 768->32 modality GEMM tile for one wave (16 nodes), two accumulator chains
// per column tile so back-to-back v_wmma can issue.  Rows are CLAMPED (not
// predicated): invalid rows only produce values for invalid nodes which are
// never consumed, so the inner loop has no exec manipulation.
// ---------------------------------------------------------------------------
__device__ __forceinline__ void gemm768_tile(const float* __restrict__ Ar,
                                             const float* __restrict__ Wm,
                                             const float* __restrict__ bm,
                                             int wave, int lrow, int hi, int mod,
                                             float (*sm_m)[4][32])
{
  v8f c0a = {}, c0b = {}, c1a = {}, c1b = {};
  const float* Wp = Wm + 2 * hi * 32 + lrow;   // column base for this lane
  for (int k0 = 0; k0 < 768; k0 += 8) {
    __builtin_prefetch(Ar + k0 + 64, 0, 3);
    v2f a0 = *(const v2f*)(Ar + k0 + 2 * hi);
    v2f a1 = *(const v2f*)(Ar + k0 + 4 + 2 * hi);
    v2f b00; b00.x = Wp[k0 * 32];            b00.y = Wp[(k0 + 1) * 32];
    v2f b01; b01.x = Wp[k0 * 32 + 16];       b01.y = Wp[(k0 + 1) * 32 + 16];
    v2f b10; b10.x = Wp[(k0 + 4) * 32];      b10.y = Wp[(k0 + 5) * 32];
    v2f b11; b11.x = Wp[(k0 + 4) * 32 + 16]; b11.y = Wp[(k0 + 5) * 32 + 16];
    c0a = wmma4(a0, b00, c0a);
    c1a = wmma4(a0, b01, c1a);
    c0b = wmma4(a1, b10, c0b);
    c1b = wmma4(a1, b11, c1b);
  }
#pragma unroll
  for (int r = 0; r < 8; ++r) {
    int nl = wave * 16 + r + 8 * hi;
    sm_m[nl][mod][lrow]      = leaky(c0a[r] + c0b[r] + bm[lrow]);
    sm_m[nl][mod][16 + lrow] = leaky(c1a[r] + c1b[r] + bm[16 + lrow]);
  }
}

// ---------------------------------------------------------------------------
// Fused per-node modality pipeline (desc/tweet WMMA, num/cat scalar, inv/spec,
// 4-token channel attention, residual+LayerNorm, token mean, aux losses).
// Block = 64 threads (2 waves), 32 nodes per block.
// ---------------------------------------------------------------------------
__global__ __launch_bounds__(64)
void modality_kernel(const float* __restrict__ desc, const float* __restrict__ tw,
                     const float* __restrict__ nump, const float* __restrict__ catp,
                     const float* __restrict__ Wd, const float* __restrict__ bd,
                     const float* __restrict__ Wt, const float* __restrict__ bt,
                     const float* __restrict__ Wn, const float* __restrict__ bn,
                     const float* __restrict__ Wc, const float* __restrict__ bc,
                     const float* __restrict__ Winv, const float* __restrict__ binv,
                     const float* __restrict__ Wsp, const float* __restrict__ bsp,
                     const float* __restrict__ Wqkv, const float* __restrict__ bqkv,
                     const float* __restrict__ Wao, const float* __restrict__ bao,
                     const float* __restrict__ lng, const float* __restrict__ lnb,
                     float* __restrict__ chmean, float* __restrict__ gacc, int N)
{
  __shared__ float sm_m[32][4][32];     // modality features
  __shared__ float sm_ch[32][4][64];    // inv|spec channels, later h = att+ch
  __shared__ float sm_att[8][4][64];    // per-pass attention (pre-projection)
  __shared__ float sm_stats[8][4][2];   // LN mu / rstd
  __shared__ float sm_aux[2];

  const int tid  = threadIdx.x;
  const int wave = tid >> 5;
  const int lane = tid & 31;
  const int lrow = lane & 15;
  const int hi   = lane >> 4;
  const int nblk = blockIdx.x * 32;

  if (tid < 2) sm_aux[tid] = 0.f;

  // ---- Phase A: WMMA GEMMs for desc & tweet ----
  {
    int rowg = nblk + wave * 16 + lrow;
    if (rowg >= N) rowg = N - 1;                 // clamp: no exec masking
    gemm768_tile(desc + (size_t)rowg * 768, Wd, bd, wave, lrow, hi, 0, sm_m);
    gemm768_tile(tw   + (size_t)rowg * 768, Wt, bt, wave, lrow, hi, 1, sm_m);
    // num (K=5) and cat (K=3): scalar, 32 lanes = 16 nodes x 2 halves
    {
      const int nl = wave * 16 + lrow;
      const int gn = nblk + nl;
      if (gn < N) {
        const float* xr = nump + (size_t)gn * 5;
        const float* cr = catp + (size_t)gn * 3;
        for (int cix = 0; cix < 16; ++cix) {
          int col = hi * 16 + cix;
          float s = bn[col];
          for (int kk = 0; kk < 5; ++kk) s += xr[kk] * Wn[kk * 32 + col];
          sm_m[nl][2][col] = leaky(s);
          float s2 = bc[col];
          for (int kk = 0; kk < 3; ++kk) s2 += cr[kk] * Wc[kk * 32 + col];
          sm_m[nl][3][col] = leaky(s2);
        }
      }
    }
  }
  __syncthreads();

  // ---- Phase B: invariant / specific projectors (32x32 per modality) ----
  {
    const int nl = tid >> 1;
    const int half = tid & 1;
    const int gn = nblk + nl;
    if (gn < N) {
      for (int mod = 0; mod < 4; ++mod) {
        for (int cc = 0; cc < 16; ++cc) {
          int col = half * 16 + cc;
          float si = binv[mod * 32 + col];
          float ss = bsp[mod * 32 + col];
          for (int b = 0; b < 32; ++b) {
            float xv = sm_m[nl][mod][b];
            si += xv * Winv[mod * 1024 + b * 32 + col];
            ss += xv * Wsp[mod * 1024 + b * 32 + col];
          }
          sm_ch[nl][mod][col]      = tanhf(si);
          sm_ch[nl][mod][32 + col] = leaky(ss);
        }
      }
    }
  }
  __syncthreads();

  // ---- Phase B2: aux losses (inv variance + spec pairwise |cos|) ----
  if (tid < 32) {
    const int nl = tid;
    const int gn = nblk + nl;
    if (gn < N) {
      float invsum = 0.f;
      for (int c = 0; c < 32; ++c) {
        float v0 = sm_ch[nl][0][c], v1 = sm_ch[nl][1][c];
        float v2 = sm_ch[nl][2][c], v3 = sm_ch[nl][3][c];
        float ctr = 0.25f * (v0 + v1 + v2 + v3);
        float d0 = v0 - ctr, d1 = v1 - ctr, d2 = v2 - ctr, d3 = v3 - ctr;
        invsum += d0 * d0 + d1 * d1 + d2 * d2 + d3 * d3;
      }
      float nrm[4];
      for (int m2 = 0; m2 < 4; ++m2) {
        float s = 0.f;
        for (int c = 0; c < 32; ++c) { float v = sm_ch[nl][m2][32 + c]; s += v * v; }
        nrm[m2] = sqrtf(s);
      }
      float ov = 0.f;
      for (int i = 0; i < 4; ++i)
        for (int j = i + 1; j < 4; ++j) {
          float dot = 0.f;
          for (int c = 0; c < 32; ++c) dot += sm_ch[nl][i][32 + c] * sm_ch[nl][j][32 + c];
          ov += fabsf(dot) / fmaxf(nrm[i] * nrm[j], 1e-8f);
        }
      atomicAdd(&sm_aux[0], invsum);
      atomicAdd(&sm_aux[1], ov);
    }
  }

  // ---- Phase C: channel attention + residual + LN + token mean ----
  for (int pass = 0; pass < 4; ++pass) {
    { // per (node, head) qkv + softmax + att  (8 nodes x 8 heads = 64 threads)
      const int nl = pass * 8 + (tid >> 3);
      const int h  = tid & 7;
      const int gn = nblk + nl;
      if (gn < N) {
        float kk[4][8];
        for (int t2 = 0; t2 < 4; ++t2)
          for (int d = 0; d < 8; ++d) {
            int col = 64 + h * 8 + d;
            float s = bqkv[col];
            for (int c = 0; c < 64; ++c) s += sm_ch[nl][t2][c] * Wqkv[c * 192 + col];
            kk[t2][d] = s;
          }
        float sc[4][4];
        for (int ti = 0; ti < 4; ++ti) {
          float qv[8];
          for (int d = 0; d < 8; ++d) {
            int col = h * 8 + d;
            float s = bqkv[col];
            for (int c = 0; c < 64; ++c) s += sm_ch[nl][ti][c] * Wqkv[c * 192 + col];
            qv[d] = s;
          }
          for (int tj = 0; tj < 4; ++tj) {
            float s = 0.f;
            for (int d = 0; d < 8; ++d) s += qv[d] * kk[tj][d];
            sc[ti][tj] = s * 0.35355339059327373f;  // 1/sqrt(8)
          }
        }
        for (int ti = 0; ti < 4; ++ti) {
          float mx = sc[ti][0];
          for (int tj = 1; tj < 4; ++tj) mx = fmaxf(mx, sc[ti][tj]);
          float sum = 0.f;
          for (int tj = 0; tj < 4; ++tj) { sc[ti][tj] = __expf(sc[ti][tj] - mx); sum += sc[ti][tj]; }
          float inv = 1.f / sum;
          for (int tj = 0; tj < 4; ++tj) sc[ti][tj] *= inv;
        }
        float vv[4][8];
        for (int t2 = 0; t2 < 4; ++t2)
          for (int d = 0; d < 8; ++d) {
            int col = 128 + h * 8 + d;
            float s = bqkv[col];
            for (int c = 0; c < 64; ++c) s += sm_ch[nl][t2][c] * Wqkv[c * 192 + col];
            vv[t2][d] = s;
          }
        for (int ti = 0; ti < 4; ++ti)
          for (int d = 0; d < 8; ++d) {
            float s = 0.f;
            for (int tj = 0; tj < 4; ++tj) s += sc[ti][tj] * vv[tj][d];
            sm_att[tid >> 3][ti][h * 8 + d] = s;
          }
      }
    }
    __syncthreads();
    { // output projection + residual into sm_ch (in place)
      const int pair = tid >> 1;
      const int ln = pair >> 2, tok = pair & 3, halfc = tid & 1;
      const int nl = pass * 8 + ln;
      const int gn = nblk + nl;
      if (gn < N) {
        for (int cc = 0; cc < 32; ++cc) {
          int c = halfc * 32 + cc;
          float s = bao[c];
          for (int d = 0; d < 64; ++d) s += sm_att[ln][tok][d] * Wao[d * 64 + c];
          sm_ch[nl][tok][c] += s;
        }
      }
    }
    __syncthreads();
    if (tid < 32) {  // LN stats per (node, token)
      const int ln = tid >> 2, tok = tid & 3;
      const int nl = pass * 8 + ln;
      const int gn = nblk + nl;
      if (gn < N) {
        float mu = 0.f;
        for (int c = 0; c < 64; ++c) mu += sm_ch[nl][tok][c];
        mu *= (1.f / 64.f);
        float var = 0.f;
        for (int c = 0; c < 64; ++c) { float d2 = sm_ch[nl][tok][c] - mu; var += d2 * d2; }
        var *= (1.f / 64.f);
        sm_stats[ln][tok][0] = mu;
        sm_stats[ln][tok][1] = rsqrtf(var + 1e-5f);
      }
    }
    __syncthreads();
    { // normalize + token mean -> chmean
      const int ln = tid >> 3, h = tid & 7;
      const int nl = pass * 8 + ln;
      const int gn = nblk + nl;
      if (gn < N) {
        for (int d = 0; d < 8; ++d) {
          int c = h * 8 + d;
          float s = 0.f;
          for (int tok = 0; tok < 4; ++tok) {
            float nv = (sm_ch[nl][tok][c] - sm_stats[ln][tok][0]) * sm_stats[ln][tok][1];
            s += nv * lng[c] + lnb[c];
          }
          chmean[(size_t)gn * 64 + c] = 0.25f * s;
        }
      }
    }
    __syncthreads();
  }
  if (tid == 0) { atomicAdd(&gacc[0], sm_aux[0]); atomicAdd(&gacc[1], sm_aux[1]); }
}

// ---------------------------------------------------------------------------
// Block WMMA GEMM, O = 128: Y[nrows x 128] = act(X[nrows x K] @ W[K x 128] + b)
// Block = 256 threads (8 waves): tile 16 rows x 128 cols, one 16x16 tile/wave.
// K chunked by 32: A chunk (16x32) and B chunk (32x128, padded) staged in LDS,
// each wave runs 8 WMMAs per chunk on two independent accumulator chains.
// ---------------------------------------------------------------------------
__global__ __launch_bounds__(256)
void gemm_block_kernel(const float* __restrict__ X, const float* __restrict__ W,
                       const float* __restrict__ bias, float* __restrict__ Y,
                       int nrows, int K, int act)
{
  __shared__ float smA[32][16];    // [k][row], conflict-free b32 reads
  __shared__ float smB[32][132];   // [k][col] padded (+4) to split bank sets

  const int tid  = threadIdx.x;
  const int wave = tid >> 5;
  const int lane = tid & 31;
  const int lrow = lane & 15, hi = lane >> 4;
  const int row0 = blockIdx.x * 16;
  const int col0 = wave * 16;

  // cooperative-load indices (constant across chunks)
  const int ar = tid >> 4;                 // 0..15 row
  const int ak = (tid & 15) * 2;           // even k within chunk
  int arow = row0 + ar; if (arow >= nrows) arow = nrows - 1;   // clamp
  const float* Abase = X + (size_t)arow * K + ak;
  const int bk = tid >> 3;                 // 0..31 k within chunk
  const int bc = (tid & 7) * 16;           // col
  const float* Bbase = W + (size_t)bk * 128 + bc;

  v8f acc0 = {}, acc1 = {};
  for (int kc = 0; kc < K; kc += 32) {
    { // A chunk: 16x32, each thread one v2f
      v2f t = *(const v2f*)(Abase + kc);
      smA[ak][ar] = t.x;
      smA[ak + 1][ar] = t.y;
    }
    { // B chunk: 32x128, each thread 16 contiguous floats (4x float4)
      const float4* src = (const float4*)(Bbase + (size_t)kc * 128);
      float4* dst = (float4*)(&smB[bk][bc]);
      dst[0] = src[0]; dst[1] = src[1]; dst[2] = src[2]; dst[3] = src[3];
    }
    __syncthreads();
#pragma unroll
    for (int kk = 0; kk < 32; kk += 8) {
      const int ka0 = kk + 2 * hi;
      const int ka1 = kk + 4 + 2 * hi;
      v2f a0; a0.x = smA[ka0][lrow];     a0.y = smA[ka0 + 1][lrow];
      v2f b0; b0.x = smB[ka0][col0 + lrow]; b0.y = smB[ka0 + 1][col0 + lrow];
      v2f a1; a1.x = smA[ka1][lrow];     a1.y = smA[ka1 + 1][lrow];
      v2f b1; b1.x = smB[ka1][col0 + lrow]; b1.y = smB[ka1 + 1][col0 + lrow];
      acc0 = wmma4(a0, b0, acc0);
      acc1 = wmma4(a1, b1, acc1);
    }
    __syncthreads();
  }
  const int c = col0 + lrow;
#pragma unroll
  for (int r = 0; r < 8; ++r) {
    int rg = row0 + r + 8 * hi;
    if (rg < nrows) {
      float v = acc0[r] + acc1[r] + bias[c];
      if (act) v = leaky(v);
      Y[(size_t)rg * 128 + c] = v;
    }
  }
}

// ---------------------------------------------------------------------------
// Small helper kernels
// ---------------------------------------------------------------------------
__global__ void zero_kernel(float* p, int n) {
  int i = blockIdx.x * 256 + threadIdx.x;
  if (i < n) p[i] = 0.f;
}

__global__ void init_softmax_kernel(float* amax, float* denom, int n) {
  int i = blockIdx.x * 256 + threadIdx.x;
  if (i < n) { amax[i] = -3.4028234663852886e38f; denom[i] = 0.f; }
}

__global__ void relproj_kernel(const float* __restrict__ rel, const float* __restrict__ We,
                               float* __restrict__ eproj) {
  int t = threadIdx.x;           // 256 threads: r in {0,1}, c in [0,128)
  int r = t >> 7, c = t & 127;
  float s = 0.f;
  for (int kk = 0; kk < 128; ++kk) s += rel[r * 128 + kk] * We[kk * 128 + c];
  eproj[t] = s;
}

__global__ void edge_alpha_kernel(const int* __restrict__ eidx, const int* __restrict__ etype,
                                  const float* __restrict__ q, const float* __restrict__ k,
                                  const float* __restrict__ eproj, float* __restrict__ exbuf,
                                  float* __restrict__ amax, int E)
{
  int idx = blockIdx.x * 256 + threadIdx.x;
  if (idx >= E * 8) return;
  int e = idx >> 3, h = idx & 7;
  int s = eidx[e], d = eidx[E + e];
  int t = etype[e]; t = t < 0 ? 0 : (t > 1 ? 1 : t);
  const float* qd = q + (size_t)d * 128 + h * 16;
  const float* ks = k + (size_t)s * 128 + h * 16;
  const float* ep = eproj + t * 128 + h * 16;
  float acc = 0.f;
#pragma unroll
  for (int c = 0; c < 16; ++c) acc += qd[c] * (ks[c] + ep[c]);
  acc *= 0.25f;   // 1/sqrt(16)
  exbuf[idx] = acc;
  atomicMaxF(&amax[d * 8 + h], acc);
}

__global__ void edge_expsum_kernel(const int* __restrict__ eidx, float* __restrict__ exbuf,
                                   const float* __restrict__ amax, float* __restrict__ denom, int E)
{
  int idx = blockIdx.x * 256 + threadIdx.x;
  if (idx >= E * 8) return;
  int e = idx >> 3, h = idx & 7;
  int d = eidx[E + e];
  float ex = __expf(exbuf[idx] - amax[d * 8 + h]);
  exbuf[idx] = ex;
  atomicAdd(&denom[d * 8 + h], ex);
}

__global__ void edge_scatter_kernel(const int* __restrict__ eidx, const int* __restrict__ etype,
                                    const float* __restrict__ v, const float* __restrict__ eproj,
                                    const float* __restrict__ exbuf, const float* __restrict__ denom,
                                    float* __restrict__ xout, int E)
{
  int idx = blockIdx.x * 256 + threadIdx.x;
  if (idx >= E * 8) return;
  int e = idx >> 3, h = idx & 7;
  int s = eidx[e], d = eidx[E + e];
  int t = etype[e]; t = t < 0 ? 0 : (t > 1 ? 1 : t);
  float a = exbuf[idx] / (denom[d * 8 + h] + 1e-16f);
  const float* vs = v + (size_t)s * 128 + h * 16;
  const float* ep = eproj + t * 128 + h * 16;
  float* od = xout + (size_t)d * 128 + h * 16;
#pragma unroll
  for (int c = 0; c < 16; ++c) atomicAdd(&od[c], (vs[c] + ep[c]) * a);
}

__global__ void leaky_kernel(const float* __restrict__ in, float* __restrict__ out, int n) {
  int i = blockIdx.x * 256 + threadIdx.x;
  if (i < n) out[i] = leaky(in[i]);
}

__global__ void head_kernel(const float* __restrict__ x, const float* __restrict__ Wh,
                            const float* __restrict__ bh, float* __restrict__ out, int N)
{
  int n = blockIdx.x * 256 + threadIdx.x;
  if (n < N) {
    const float* xr = x + (size_t)n * 128;
    float s0 = bh[0], s1 = bh[1];
    for (int kk = 0; kk < 128; ++kk) { float xv = xr[kk]; s0 += xv * Wh[kk * 2]; s1 += xv * Wh[kk * 2 + 1]; }
    out[n * 2] = s0; out[n * 2 + 1] = s1;
  }
}

__global__ void aux_kernel(const float* __restrict__ acc, float* __restrict__ out, int N) {
  if (threadIdx.x == 0 && blockIdx.x == 0) {
    float invl = acc[0] / (float)(N * 4 * 32);
    float ov   = acc[1] / ((float)N * 6.0f);
    out[2 * N] = 0.1f * (invl + 0.5f * ov);
  }
}

// ---------------------------------------------------------------------------
extern "C" void kernel_launch(void* const* d_in, const int* in_sizes, int n_in,
                              void* d_out, int out_size, void* d_ws, size_t ws_size,
                              hipStream_t stream)
{
  const int N = in_sizes[0] / 768;
  const int E = in_sizes[5];

  const float* desc  = (const float*)d_in[0];
  const float* tw    = (const float*)d_in[1];
  const float* nump  = (const float*)d_in[2];
  const float* catp  = (const float*)d_in[3];
  const int*   eidx  = (const int*)d_in[4];
  const int*   etype = (const int*)d_in[5];
  const float* W_desc = (const float*)d_in[6];  const float* b_desc = (const float*)d_in[7];
  const float* W_tw   = (const float*)d_in[8];  const float* b_tw   = (const float*)d_in[9];
  const float* W_num  = (const float*)d_in[10]; const float* b_num  = (const float*)d_in[11];
  const float* W_cat  = (const float*)d_in[12]; const float* b_cat  = (const float*)d_in[13];
  const float* W_inv  = (const float*)d_in[14]; const float* b_inv  = (const float*)d_in[15];
  const float* W_spec = (const float*)d_in[16]; const float* b_spec = (const float*)d_in[17];
  const float* W_qkv  = (const float*)d_in[18]; const float* b_qkv  = (const float*)d_in[19];
  const float* W_ao   = (const float*)d_in[20]; const float* b_ao   = (const float*)d_in[21];
  const float* ln_g   = (const float*)d_in[22]; const float* ln_b   = (const float*)d_in[23];
  const float* W_c2h  = (const float*)d_in[24]; const float* b_c2h  = (const float*)d_in[25];
  const float* rel    = (const float*)d_in[26];
  const float* Wq[2]  = {(const float*)d_in[27], (const float*)d_in[36]};
  const float* bq[2]  = {(const float*)d_in[28], (const float*)d_in[37]};
  const float* Wk[2]  = {(const float*)d_in[29], (const float*)d_in[38]};
  const float* bk[2]  = {(const float*)d_in[30], (const float*)d_in[39]};
  const float* Wv[2]  = {(const float*)d_in[31], (const float*)d_in[40]};
  const float* bv[2]  = {(const float*)d_in[32], (const float*)d_in[41]};
  const float* We[2]  = {(const float*)d_in[33], (const float*)d_in[42]};
  const float* Wsk[2] = {(const float*)d_in[34], (const float*)d_in[43]};
  const float* bsk[2] = {(const float*)d_in[35], (const float*)d_in[44]};
  const float* W_out  = (const float*)d_in[45]; const float* b_out  = (const float*)d_in[46];
  const float* W_head = (const float*)d_in[47]; const float* b_head = (const float*)d_in[48];

  float* out = (float*)d_out;

  // workspace layout (floats)
  float* ws = (float*)d_ws;
  size_t nf = (size_t)N * 128;
  float* x    = ws;
  float* q    = x + nf;
  float* kbuf = q + nf;
  float* vbuf = kbuf + nf;
  float* xout = vbuf + nf;
  size_t er = (size_t)E * 8;
  size_t cm = (size_t)N * 64;
  if (cm > er) er = cm;
  float* ebuf  = xout + nf;      // chmean first, then per-edge alpha/exp buffer
  float* amax  = ebuf + er;
  float* denom = amax + (size_t)N * 8;
  float* eproj = denom + (size_t)N * 8;
  float* acc   = eproj + 256;

  // 1) aux accumulators
  zero_kernel<<<1, 256, 0, stream>>>(acc, 8);

  // 2) fused modality pipeline -> chmean [N,64] (in ebuf), aux partials
  modality_kernel<<<(N + 31) / 32, 64, 0, stream>>>(
      desc, tw, nump, catp, W_desc, b_desc, W_tw, b_tw, W_num, b_num, W_cat, b_cat,
      W_inv, b_inv, W_spec, b_spec, W_qkv, b_qkv, W_ao, b_ao, ln_g, ln_b,
      ebuf, acc, N);

  const int gblocks = (N + 15) / 16;

  // 3) fused = leaky(chmean @ W_c2h + b)  [N,64]->[N,128]
  gemm_block_kernel<<<gblocks, 256, 0, stream>>>(ebuf, W_c2h, b_c2h, x, N, 64, 1);

  // 4) two TransformerConv layers
  const int eblocks = (E * 8 + 255) / 256;
  for (int L = 0; L < 2; ++L) {
    relproj_kernel<<<1, 256, 0, stream>>>(rel, We[L], eproj);   // e = rel_emb @ W_e (2x128)
    gemm_block_kernel<<<gblocks, 256, 0, stream>>>(x, Wq[L],  bq[L],  q,    N, 128, 0);
    gemm_block_kernel<<<gblocks, 256, 0, stream>>>(x, Wk[L],  bk[L],  kbuf, N, 128, 0);
    gemm_block_kernel<<<gblocks, 256, 0, stream>>>(x, Wv[L],  bv[L],  vbuf, N, 128, 0);
    gemm_block_kernel<<<gblocks, 256, 0, stream>>>(x, Wsk[L], bsk[L], xout, N, 128, 0); // skip path
    init_softmax_kernel<<<(N * 8 + 255) / 256, 256, 0, stream>>>(amax, denom, N * 8);
    edge_alpha_kernel  <<<eblocks, 256, 0, stream>>>(eidx, etype, q, kbuf, eproj, ebuf, amax, E);
    edge_expsum_kernel <<<eblocks, 256, 0, stream>>>(eidx, ebuf, amax, denom, E);
    edge_scatter_kernel<<<eblocks, 256, 0, stream>>>(eidx, etype, vbuf, eproj, ebuf, denom, xout, E);
    if (L == 0)
      leaky_kernel<<<((int)nf + 255) / 256, 256, 0, stream>>>(xout, x, (int)nf);
  }

  // 5) x = leaky(conv2_out @ W_out + b_out) -> q (temp), then head
  gemm_block_kernel<<<gblocks, 256, 0, stream>>>(xout, W_out, b_out, q, N, 128, 1);
  head_kernel<<<(N + 255) / 256, 256, 0, stream>>>(q, W_head, b_head, out, N);
  aux_kernel<<<1, 64, 0, stream>>>(acc, out, N);
}